// MambaBlock_43138651521745
// MI455X (gfx1250) — compile-verified
//
#include <hip/hip_runtime.h>
#include <hip/hip_bf16.h>
#include <math.h>

// ---------------------------------------------------------------------------
// Mamba block for MI455X (gfx1250, wave32, WMMA).
// fp32 end-to-end using V_WMMA_F32_16X16X4_F32 to match the fp32 reference.
// A-panel staging uses GLOBAL_LOAD_ASYNC_TO_LDS (ASYNCcnt) when available.
// ---------------------------------------------------------------------------

typedef __attribute__((ext_vector_type(2))) float v2f;
typedef __attribute__((ext_vector_type(8))) float v8f;
typedef __attribute__((ext_vector_type(4))) int   v4i;

#define D_MODEL 1024
#define D_STATE 16
#define D_CONV  4
#define D_INNER 2048
#define BATCH   4
#define SEQ     1024
#define ROWS    (BATCH * SEQ)          // 4096 token rows
#define LN_EPS  1e-5f

#if defined(__has_builtin)
#if __has_builtin(__builtin_amdgcn_global_load_async_to_lds_b128) && \
    __has_builtin(__builtin_amdgcn_s_wait_asynccnt)
#define USE_ASYNC_LDS 1
#endif
#endif

__device__ __forceinline__ float silu_f(float x) {
  return x / (1.0f + __expf(-x));
}
__device__ __forceinline__ float softplus_f(float x) {
  return (x > 20.0f) ? x : log1pf(__expf(x));
}

__device__ __forceinline__ v8f wmma_f32(v2f a, v2f b, v8f c) {
  return __builtin_amdgcn_wmma_f32_16x16x4_f32(
      /*neg_a=*/false, a, /*neg_b=*/false, b,
      /*c_mod=*/(short)0, c, /*reuse_a=*/false, /*reuse_b=*/false);
}

// ---------------------------------------------------------------------------
// Tiled fp32 WMMA GEMM: C[M,N] = epilogue(A[M,K] @ W[K,N] + bias[N])
//
// Block (256 thr, 8 waves) computes a 16 x 512 C panel:
//   - A panel 16xKC staged in LDS, via async global->LDS copy when the
//     toolchain exposes it (b128: 256 lanes x 16B = full 4KB panel in one
//     instruction), else coalesced VGPR staging. Row stride padded to 66
//     so per-lane fragment reads are bank-conflict-free ds_load_b64.
//   - Each wave owns a 16x64 strip = 4 N-tiles, reusing its A fragment
//     across 4 wmma per K-step (B loads are lane-coalesced dwords).
//
// Fragment layout (ISA 7.12.2, wave32, V_WMMA_F32_16X16X4_F32):
//   A (16x4): lane r(0-15)=row; half=lane>>4 picks K pair {2h, 2h+1}
//   B (4x16): b.x = W[k0+2h][n0+r], b.y = W[k0+2h+1][n0+r]
//   C/D     : vgpr v, lane -> C[m0 + v + 8*half][n0 + r]
//
// MODE: 0 = bias only, 1 = softplus, 2 = add residual (addp[M,N])
// ---------------------------------------------------------------------------
#define KC      64
#define LDS_STR (KC + 2)   // 66: bank-conflict-free (stride%64 == 2)

template <int MODE>
__global__ __launch_bounds__(256)
void gemm_wmma_tiled(const float* __restrict__ A, const float* __restrict__ W,
                     const float* __restrict__ bias, const float* __restrict__ addp,
                     float* __restrict__ C, int M, int N, int K) {
  __shared__ float sA[16 * LDS_STR];

  const int lane = threadIdx.x & 31;
  const int wave = threadIdx.x >> 5;
  const int half = lane >> 4;
  const int r    = lane & 15;

  const int nblk  = N >> 9;                         // N / 512
  const int m0    = (blockIdx.x / nblk) << 4;
  const int nbase = ((blockIdx.x % nblk) << 9) + (wave << 6);

  v8f acc0 = {}, acc1 = {}, acc2 = {}, acc3 = {};

  const float* Wb = W + (size_t)(2 * half) * N + (nbase + r);

  for (int kb = 0; kb < K; kb += KC) {
    __syncthreads();   // protect sA from previous iteration's readers
#if USE_ASYNC_LDS
    {
      // One b128 per thread: 16 rows x 64 floats = 4KB panel, direct to LDS.
      const int rr = threadIdx.x >> 4;          // row 0..15
      const int cc = (threadIdx.x & 15) << 2;   // col 0,4,..,60
      __builtin_amdgcn_global_load_async_to_lds_b128(
          (v4i*)(A + (size_t)(m0 + rr) * K + kb + cc),
          (v4i*)(&sA[rr * LDS_STR + cc]),
          0, 0);
      __builtin_amdgcn_s_wait_asynccnt(0);
    }
#else
    // Cooperative, coalesced load of A panel: 16 rows x KC floats.
#pragma unroll
    for (int i = threadIdx.x; i < 16 * KC; i += 256) {
      const int rr = i >> 6;          // KC == 64
      const int cc = i & (KC - 1);
      sA[rr * LDS_STR + cc] = A[(size_t)(m0 + rr) * K + kb + cc];
    }
#endif
    __syncthreads();

#pragma unroll
    for (int k0 = 0; k0 < KC; k0 += 4) {
      v2f a;
      a.x = sA[r * LDS_STR + k0 + 2 * half];
      a.y = sA[r * LDS_STR + k0 + 2 * half + 1];

      const float* Wk = Wb + (size_t)(kb + k0) * N;
      v2f b0, b1, b2, b3;
      b0.x = Wk[0];            b0.y = Wk[(size_t)N + 0];
      b1.x = Wk[16];           b1.y = Wk[(size_t)N + 16];
      b2.x = Wk[32];           b2.y = Wk[(size_t)N + 32];
      b3.x = Wk[48];           b3.y = Wk[(size_t)N + 48];
      // Prefetch B rows 2 K-steps (8 k-rows) ahead (speculative, OOB-safe).
      __builtin_prefetch(&Wk[(size_t)8 * N], 0, 1);

      acc0 = wmma_f32(a, b0, acc0);
      acc1 = wmma_f32(a, b1, acc1);
      acc2 = wmma_f32(a, b2, acc2);
      acc3 = wmma_f32(a, b3, acc3);
    }
  }

  // Epilogue: 4 N-tiles per wave.
#pragma unroll
  for (int t = 0; t < 4; ++t) {
    const v8f acc = (t == 0) ? acc0 : (t == 1) ? acc1 : (t == 2) ? acc2 : acc3;
    const int col = nbase + (t << 4) + r;
    const float bv = bias[col];
#pragma unroll
    for (int v = 0; v < 8; ++v) {
      const int row = m0 + v + 8 * half;
      float val = acc[v] + bv;
      if (MODE == 1) val = softplus_f(val);
      if (MODE == 2) val += addp[(size_t)row * N + col];
      C[(size_t)row * N + col] = val;
    }
  }
}

// ---------------------------------------------------------------------------
// Small-N fp32 WMMA GEMM (x_proj, N=32): one wave -> one 16x16 tile.
// ---------------------------------------------------------------------------
__global__ __launch_bounds__(256)
void gemm_wmma_small(const float* __restrict__ A, const float* __restrict__ W,
                     const float* __restrict__ bias, float* __restrict__ C,
                     int M, int N, int K) {
  const int lane = threadIdx.x & 31;
  const int wave = threadIdx.x >> 5;
  const int half = lane >> 4;
  const int r    = lane & 15;

  const int tiles_n = N >> 4;
  const int tile = blockIdx.x * 8 + wave;
  if (tile >= (M >> 4) * tiles_n) return;   // uniform per wave
  const int m0 = (tile / tiles_n) << 4;
  const int n0 = (tile % tiles_n) << 4;

  const float* Arow = A + (size_t)(m0 + r) * K + 2 * half;
  const float* Wcol = W + (size_t)(2 * half) * N + (n0 + r);

  v8f acc = {};
#pragma unroll 4
  for (int k0 = 0; k0 < K; k0 += 4) {
    v2f a, b;
    a.x = Arow[k0];
    a.y = Arow[k0 + 1];
    b.x = Wcol[(size_t)k0 * N];
    b.y = Wcol[(size_t)(k0 + 1) * N];
    acc = wmma_f32(a, b, acc);
  }

  const float bv = bias[n0 + r];
#pragma unroll
  for (int v = 0; v < 8; ++v)
    C[(size_t)(m0 + v + 8 * half) * N + (n0 + r)] = acc[v] + bv;
}

// ---------------------------------------------------------------------------
// u = silu( depthwise_causal_conv( silu(xi) ) ), xi = xr[:, :, 0:D_INNER]
// ---------------------------------------------------------------------------
__global__ __launch_bounds__(256)
void conv_silu_kernel(const float* __restrict__ xr, const float* __restrict__ cw,
                      const float* __restrict__ cb, float* __restrict__ u) {
  const int idx = blockIdx.x * blockDim.x + threadIdx.x;
  if (idx >= ROWS * D_INNER) return;
  const int d = idx % D_INNER;
  const int l = (idx / D_INNER) % SEQ;
  const int b = idx / (D_INNER * SEQ);

  const float* xb = xr + (size_t)b * SEQ * (2 * D_INNER) + d;
  float acc = cb[d];
#pragma unroll
  for (int j = 0; j < D_CONV; ++j) {
    const int ls = l - (D_CONV - 1) + j;
    const float xv = (ls >= 0) ? silu_f(xb[(size_t)ls * (2 * D_INNER)]) : 0.0f;
    acc += cw[d * D_CONV + j] * xv;
  }
  u[idx] = silu_f(acc);
}

// ---------------------------------------------------------------------------
// Selective scan. Thread owns channel d; 16-wide state in registers.
// B/C staged in LDS in 64-step chunks. Fuses +u*D and *silu(res).
// ---------------------------------------------------------------------------
#define SCAN_TPB 256
#define CHUNK    64
__global__ __launch_bounds__(SCAN_TPB)
void scan_kernel(const float* __restrict__ u, const float* __restrict__ dt,
                 const float* __restrict__ bc, const float* __restrict__ xr,
                 const float* __restrict__ A_log, const float* __restrict__ Dp_,
                 float* __restrict__ yz) {
  __shared__ float sBC[CHUNK * 32];
  const int b = blockIdx.y;
  const int d = blockIdx.x * SCAN_TPB + threadIdx.x;

  float A[D_STATE], h[D_STATE];
#pragma unroll
  for (int n = 0; n < D_STATE; ++n) {
    A[n] = -__expf(A_log[d * D_STATE + n]);
    h[n] = 0.0f;
  }
  const float Dp = Dp_[d];

  const size_t baseLD = (size_t)b * SEQ * D_INNER;
  const size_t baseBC = (size_t)b * SEQ * 32;
  const size_t baseXR = (size_t)b * SEQ * (2 * D_INNER);

  for (int l0 = 0; l0 < SEQ; l0 += CHUNK) {
    for (int i = threadIdx.x; i < CHUNK * 32; i += SCAN_TPB)
      sBC[i] = bc[baseBC + (size_t)l0 * 32 + i];
    __syncthreads();

    for (int li = 0; li < CHUNK; ++li) {
      const int l = l0 + li;
      const float dtv = dt[baseLD + (size_t)l * D_INNER + d];
      const float uv  = u [baseLD + (size_t)l * D_INNER + d];
      const float* Bv = &sBC[li * 32];
      const float* Cv = Bv + 16;
      const float dtu = dtv * uv;
      float y = 0.0f;
#pragma unroll
      for (int n = 0; n < D_STATE; ++n) {
        const float ad = __expf(dtv * A[n]);     // exp(dt * A)
        h[n] = ad * h[n] + dtu * Bv[n];          // h = a*h + dt*B*x
        y += h[n] * Cv[n];                       // y = <h, C>
      }
      y += uv * Dp;
      const float resv = xr[baseXR + (size_t)l * (2 * D_INNER) + D_INNER + d];
      yz[baseLD + (size_t)l * D_INNER + d] = y * silu_f(resv);
    }
    __syncthreads();
  }
}

// ---------------------------------------------------------------------------
// In-place LayerNorm over last dim (1024). One 256-thread block per row.
// ---------------------------------------------------------------------------
__global__ __launch_bounds__(256)
void layernorm_kernel(float* __restrict__ z, const float* __restrict__ g,
                      const float* __restrict__ beta) {
  __shared__ float ssum[256], ssq[256];
  float* zr = z + (size_t)blockIdx.x * D_MODEL;

  float lsum = 0.0f, lsq = 0.0f;
  for (int i = threadIdx.x; i < D_MODEL; i += 256) {
    const float v = zr[i];
    lsum += v;
    lsq  += v * v;
  }
  ssum[threadIdx.x] = lsum;
  ssq[threadIdx.x]  = lsq;
  __syncthreads();
  for (int s = 128; s > 0; s >>= 1) {
    if (threadIdx.x < s) {
      ssum[threadIdx.x] += ssum[threadIdx.x + s];
      ssq[threadIdx.x]  += ssq[threadIdx.x + s];
    }
    __syncthreads();
  }
  const float mu  = ssum[0] * (1.0f / D_MODEL);
  const float var = ssq[0] * (1.0f / D_MODEL) - mu * mu;
  const float inv = rsqrtf(var + LN_EPS);
  for (int i = threadIdx.x; i < D_MODEL; i += 256)
    zr[i] = (zr[i] - mu) * inv * g[i] + beta[i];
}

// ---------------------------------------------------------------------------
// Launch
// ---------------------------------------------------------------------------
extern "C" void kernel_launch(void* const* d_in, const int* in_sizes, int n_in,
                              void* d_out, int out_size, void* d_ws, size_t ws_size,
                              hipStream_t stream) {
  const float* x      = (const float*)d_in[0];
  const float* W_in   = (const float*)d_in[1];
  const float* b_in   = (const float*)d_in[2];
  const float* conv_w = (const float*)d_in[3];
  const float* conv_b = (const float*)d_in[4];
  const float* W_xp   = (const float*)d_in[5];
  const float* b_xp   = (const float*)d_in[6];
  const float* W_dt   = (const float*)d_in[7];
  const float* b_dt   = (const float*)d_in[8];
  const float* A_log  = (const float*)d_in[9];
  const float* D_par  = (const float*)d_in[10];
  const float* W_out  = (const float*)d_in[11];
  const float* b_out  = (const float*)d_in[12];
  const float* ln_g   = (const float*)d_in[13];
  const float* ln_b   = (const float*)d_in[14];
  float* out = (float*)d_out;

  // Workspace layout (floats)
  float* ws  = (float*)d_ws;
  float* xr  = ws;                                   // ROWS * 4096
  float* u   = xr + (size_t)ROWS * 2 * D_INNER;      // ROWS * 2048
  float* dtb = u  + (size_t)ROWS * D_INNER;          // ROWS * 2048
  float* bc  = dtb + (size_t)ROWS * D_INNER;         // ROWS * 32
  float* yz  = bc  + (size_t)ROWS * 32;              // ROWS * 2048

  // 1) xr = x @ W_in + b_in        (M=4096, N=4096, K=1024)
  {
    const int blocks = (ROWS / 16) * ((2 * D_INNER) / 512);   // 256*8 = 2048
    gemm_wmma_tiled<0><<<blocks, 256, 0, stream>>>(
        x, W_in, b_in, nullptr, xr, ROWS, 2 * D_INNER, D_MODEL);
  }
  // 2) u = silu(conv(silu(xi)))
  {
    const int n = ROWS * D_INNER;
    conv_silu_kernel<<<(n + 255) / 256, 256, 0, stream>>>(xr, conv_w, conv_b, u);
  }
  // 3) bc = u @ W_xp + b_xp        (M=4096, N=32, K=2048)
  {
    const int tiles = (ROWS / 16) * ((2 * D_STATE) / 16);     // 512
    gemm_wmma_small<<<tiles / 8, 256, 0, stream>>>(
        u, W_xp, b_xp, bc, ROWS, 2 * D_STATE, D_INNER);
  }
  // 4) dt = softplus(u @ W_dt + b_dt)   (M=4096, N=2048, K=2048)
  {
    const int blocks = (ROWS / 16) * (D_INNER / 512);         // 256*4 = 1024
    gemm_wmma_tiled<1><<<blocks, 256, 0, stream>>>(
        u, W_dt, b_dt, nullptr, dtb, ROWS, D_INNER, D_INNER);
  }
  // 5) selective scan -> yz = (ssm(u) + u*D) * silu(res)
  {
    dim3 grid(D_INNER / SCAN_TPB, BATCH);
    scan_kernel<<<grid, SCAN_TPB, 0, stream>>>(u, dtb, bc, xr, A_log, D_par, yz);
  }
  // 6) z = yz @ W_out + b_out + x   (M=4096, N=1024, K=2048) -> d_out
  {
    const int blocks = (ROWS / 16) * (D_MODEL / 512);         // 256*2 = 512
    gemm_wmma_tiled<2><<<blocks, 256, 0, stream>>>(
        yz, W_out, b_out, x, out, ROWS, D_MODEL, D_INNER);
  }
  // 7) LayerNorm in place on d_out
  layernorm_kernel<<<ROWS, 256, 0, stream>>>(out, ln_g, ln_b);
}